// DiagOUSDE_39032662786269
// MI455X (gfx1250) — compile-verified
//
#include <hip/hip_runtime.h>
#include <math.h>

typedef float v4f __attribute__((ext_vector_type(4)));
typedef float v2f __attribute__((ext_vector_type(2)));
typedef float v8f __attribute__((ext_vector_type(8)));

#define LOG2PI_F 1.8378770664093453f

static constexpr int Bv = 256;   // batch
static constexpr int Tv = 1024;  // time
static constexpr int Dv = 256;   // dims
static constexpr int SUBT   = 4;            // t-rows per wave
static constexpr int WAVES  = 8;            // waves per block (256 threads, wave32)
static constexpr int TILE_T = SUBT * WAVES; // 32 t-rows per block
static constexpr int BG     = 16;           // batches per block
static constexpr int NSUB   = Tv / SUBT;    // 256 global subtiles

__device__ __forceinline__ float softplusf(float x) {
    return (x > 20.0f) ? x : log1pf(expf(x));
}

// ---------------------------------------------------------------------------
// Kernel 1: batch-independent constant  c(t,d) = log(max(q,1e-10)) + LOG2PI
// block-reduced into Cp[1024] (deterministic, no atomics)
// ---------------------------------------------------------------------------
__global__ __launch_bounds__(256) void k_csum(const float* __restrict__ ts,
                                              const float* __restrict__ lkap,
                                              const float* __restrict__ lsig,
                                              float* __restrict__ Cp) {
    const int idx = blockIdx.x * 256 + threadIdx.x; // [0, T*D)
    const int t = idx >> 8;        // D == 256
    const int d = idx & 255;
    const float kap = softplusf(lkap[d]) + 1e-6f;
    const float sig = softplusf(lsig[d]) + 1e-6f;
    const float s2  = sig * sig;
    float q;
    if (t == 0) {
        q = s2 / (2.0f * kap);
    } else {
        const float dtv = fmaxf(ts[t * Dv + d] - ts[(t - 1) * Dv + d], 1e-6f);
        q = s2 * (1.0f - expf(-2.0f * kap * dtv)) / (2.0f * kap);
    }
    float c = logf(fmaxf(q, 1e-10f)) + LOG2PI_F;

    // wave32 reduce then combine 8 waves through LDS
    #pragma unroll
    for (int off = 16; off > 0; off >>= 1) c += __shfl_xor(c, off, 32);
    __shared__ float lds[WAVES];
    const int lane = threadIdx.x & 31, wid = threadIdx.x >> 5;
    if (lane == 0) lds[wid] = c;
    __syncthreads();
    if (threadIdx.x == 0) {
        float s = 0.0f;
        #pragma unroll
        for (int i = 0; i < WAVES; ++i) s += lds[i];
        Cp[blockIdx.x] = s;
    }
}

// ---------------------------------------------------------------------------
// Kernel 2: stream y once; accumulate diff^2 / q.
// Wave w owns t-rows [t0, t0+4); lane owns 8 d-columns; loops 16 batches.
// Cross-lane reduction via V_WMMA_F32_16X16X4_F32 ones-matrix trick.
// ---------------------------------------------------------------------------
__global__ __launch_bounds__(256) void k_main(const float* __restrict__ y,
                                              const float* __restrict__ ts,
                                              const float* __restrict__ mu,
                                              const float* __restrict__ lkap,
                                              const float* __restrict__ lsig,
                                              float* __restrict__ partial) {
    const int lane = threadIdx.x & 31;
    const int wid  = threadIdx.x >> 5;
    const int t0   = blockIdx.x * TILE_T + wid * SUBT;
    const int st   = blockIdx.x * WAVES + wid;   // global subtile id
    const int d0   = lane * 8;                   // this lane's 8 dims

    // per-dim parameters (registers)
    float muv[8], kap[8], s2[8];
    {
        v4f m0 = *(const v4f*)(mu + d0);
        v4f m1 = *(const v4f*)(mu + d0 + 4);
        #pragma unroll
        for (int j = 0; j < 4; ++j) { muv[j] = m0[j]; muv[4 + j] = m1[j]; }
        #pragma unroll
        for (int j = 0; j < 8; ++j) {
            const float kk = softplusf(lkap[d0 + j]) + 1e-6f;
            const float ss = softplusf(lsig[d0 + j]) + 1e-6f;
            kap[j] = kk; s2[j] = ss * ss;
        }
    }

    // transition coefficients for my 4 rows: A = exp(-k*dt), R = 1/max(q,1e-10)
    float A[SUBT][8], R[SUBT][8];
    {
        float tsp[8];
        if (t0 > 0) {
            v4f p0 = *(const v4f*)(ts + (size_t)(t0 - 1) * Dv + d0);
            v4f p1 = *(const v4f*)(ts + (size_t)(t0 - 1) * Dv + d0 + 4);
            #pragma unroll
            for (int j = 0; j < 4; ++j) { tsp[j] = p0[j]; tsp[4 + j] = p1[j]; }
        } else {
            #pragma unroll
            for (int j = 0; j < 8; ++j) tsp[j] = 0.0f;
        }
        #pragma unroll
        for (int k = 0; k < SUBT; ++k) {
            const int t = t0 + k;
            v4f c0 = *(const v4f*)(ts + (size_t)t * Dv + d0);
            v4f c1 = *(const v4f*)(ts + (size_t)t * Dv + d0 + 4);
            float tc[8];
            #pragma unroll
            for (int j = 0; j < 4; ++j) { tc[j] = c0[j]; tc[4 + j] = c1[j]; }
            #pragma unroll
            for (int j = 0; j < 8; ++j) {
                float a, q;
                if (t == 0) {               // wave-uniform branch
                    a = 0.0f;
                    q = s2[j] / (2.0f * kap[j]);
                } else {
                    const float dtv = fmaxf(tc[j] - tsp[j], 1e-6f);
                    a = expf(-kap[j] * dtv);
                    q = s2[j] * (1.0f - expf(-2.0f * kap[j] * dtv)) / (2.0f * kap[j]);
                }
                A[k][j] = a;
                R[k][j] = 1.0f / fmaxf(q, 1e-10f);
                tsp[j]  = tc[j];
            }
        }
    }

    const int bbase = blockIdx.y * BG;
    for (int i = 0; i < BG; ++i) {
        const int b = bbase + i;
        const float* yb = y + ((size_t)b * Tv + t0) * Dv + d0;
        __builtin_prefetch(yb + (size_t)Tv * Dv, 0, 3); // next batch's rows

        float pd[8]; // y[t-1] - mu, kept in registers across the 4 rows
        if (t0 > 0) {
            v4f p0 = *(const v4f*)(yb - Dv);
            v4f p1 = *(const v4f*)(yb - Dv + 4);
            #pragma unroll
            for (int j = 0; j < 4; ++j) { pd[j] = p0[j] - muv[j]; pd[4 + j] = p1[j] - muv[4 + j]; }
        } else {
            #pragma unroll
            for (int j = 0; j < 8; ++j) pd[j] = 0.0f; // A[0]=0 makes it unused
        }

        float acc = 0.0f;
        #pragma unroll
        for (int k = 0; k < SUBT; ++k) {
            v4f c0 = *(const v4f*)(yb + (size_t)k * Dv);
            v4f c1 = *(const v4f*)(yb + (size_t)k * Dv + 4);
            #pragma unroll
            for (int j = 0; j < 4; ++j) {
                const float e  = c0[j] - muv[j];
                const float df = e - A[k][j] * pd[j];
                acc = fmaf(df * df, R[k][j], acc);
                pd[j] = e;
            }
            #pragma unroll
            for (int j = 0; j < 4; ++j) {
                const float e  = c1[j] - muv[4 + j];
                const float df = e - A[k][4 + j] * pd[4 + j];
                acc = fmaf(df * df, R[k][4 + j], acc);
                pd[4 + j] = e;
            }
        }

        // --- cross-lane reduction via WMMA (A = lane data, B = ones) ---
        // D[m,n] = s[m] + s[m+16]; per-lane sum of 8 D VGPRs gives half-sums,
        // one shfl_xor(16) completes the 32-lane total. fp32 throughout.
        v2f av; av[0] = acc;  av[1] = 0.0f;
        v2f bv; bv[0] = 1.0f; bv[1] = 1.0f;
        v8f cz = {};
        v8f dm = __builtin_amdgcn_wmma_f32_16x16x4_f32(
            false, av, false, bv, (short)0, cz, false, false);
        float s8 = dm[0] + dm[1] + dm[2] + dm[3] + dm[4] + dm[5] + dm[6] + dm[7];
        const float tot = s8 + __shfl_xor(s8, 16, 32);
        if (lane == 0) partial[(size_t)st * Bv + b] = tot;
    }
}

// ---------------------------------------------------------------------------
// Kernel 3: deterministic final combine: out[b] = -0.5*(SC + sum_st partial)
// ---------------------------------------------------------------------------
__global__ __launch_bounds__(256) void k_final(const float* __restrict__ Cp,
                                               const float* __restrict__ partial,
                                               float* __restrict__ out) {
    __shared__ float lds[256];
    const int tid = threadIdx.x;
    lds[tid] = Cp[tid] + Cp[tid + 256] + Cp[tid + 512] + Cp[tid + 768];
    __syncthreads();
    for (int s = 128; s > 0; s >>= 1) {
        if (tid < s) lds[tid] += lds[tid + s];
        __syncthreads();
    }
    const float SC = lds[0];
    float s = 0.0f;
    for (int st = 0; st < NSUB; ++st) s += partial[(size_t)st * Bv + tid];
    out[tid] = -0.5f * (SC + s);
}

// ---------------------------------------------------------------------------
extern "C" void kernel_launch(void* const* d_in, const int* in_sizes, int n_in,
                              void* d_out, int out_size, void* d_ws, size_t ws_size,
                              hipStream_t stream) {
    const float* y  = (const float*)d_in[0]; // (B,T,D)
    const float* ts = (const float*)d_in[1]; // (T,D)
    const float* mu = (const float*)d_in[2]; // (D)
    const float* lk = (const float*)d_in[3]; // (D)
    const float* ls = (const float*)d_in[4]; // (D)
    float* out = (float*)d_out;              // (B)

    // workspace: Cp[1024] + partial[NSUB*B] = (1024 + 65536)*4 B ~ 260 KB
    float* Cp      = (float*)d_ws;
    float* partial = Cp + 1024;

    k_csum <<<dim3((Tv * Dv) / 256), 256, 0, stream>>>(ts, lk, ls, Cp);
    k_main <<<dim3(Tv / TILE_T, Bv / BG), 256, 0, stream>>>(y, ts, mu, lk, ls, partial);
    k_final<<<dim3(1), 256, 0, stream>>>(Cp, partial, out);
}